// LennardJones_36206574305302
// MI455X (gfx1250) — compile-verified
//
#include <hip/hip_runtime.h>
#include <stdint.h>

#define TPB   256          // threads per block (8 waves of 32)
#define TILE  1024         // edges per tile
// per-tile async traffic: bv = TILE*12 B = 768 x b128 = 3 per thread
//                         idx = TILE*4  B = 256 x b128 = 1 per thread
#define ASYNC_PER_TILE 4

// energy shift at cutoff: e0 = 4*((1/3)^12 - (1/3)^6)
#define E0_SHIFT (-5.4794411547e-3f)

__device__ __forceinline__ void async_b128(uint32_t lds_off, uint32_t goff, uint64_t sbase) {
    // GVS form: global_load_async_to_lds_b128 vdst(lds addr), vaddr(32b offset), saddr(64b base)
    asm volatile("global_load_async_to_lds_b128 %0, %1, %2"
                 :: "v"(lds_off), "v"(goff), "s"(sbase)
                 : "memory");
}

__device__ __forceinline__ void wait_async_le(void) {
    asm volatile("s_wait_asynccnt %0" :: "n"(ASYNC_PER_TILE) : "memory");
}
__device__ __forceinline__ void wait_async_zero(void) {
    asm volatile("s_wait_asynccnt 0x0" ::: "memory");
}

__device__ __forceinline__ void lj_edge(float bx, float by, float bz, int node,
                                        float* __restrict__ force_out, float& esum) {
    float r2  = fmaf(bx, bx, fmaf(by, by, bz * bz));
    float inv = 1.0f / r2;
    float c6  = inv * inv * inv;
    float c12 = c6 * c6;
    esum += 0.5f * (4.0f * (c12 - c6) - E0_SHIFT);
    float fs = 12.0f * (2.0f * c12 - c6) * inv;   // forces = segsum(fs * bv)
    float* f = force_out + 3 * node;
    atomicAdd(f + 0, fs * bx);
    atomicAdd(f + 1, fs * by);
    atomicAdd(f + 2, fs * bz);
}

__device__ __forceinline__ void energy_reduce_commit(float esum, float* energy_out) {
    #pragma unroll
    for (int o = 16; o > 0; o >>= 1)
        esum += __shfl_down(esum, o, 32);
    if ((threadIdx.x & 31u) == 0u)
        atomicAdd(energy_out, esum);
}

// ---------------- main tiled kernel: async global->LDS double buffering --------------
__global__ void __launch_bounds__(TPB)
lj_main(const float* __restrict__ bv, const int* __restrict__ dst,
        float* __restrict__ out, int ntiles) {
    __shared__ alignas(16) float s_bv[2][TILE * 3];
    __shared__ alignas(16) int   s_ix[2][TILE];

    const uint32_t lds_bv = (uint32_t)(uintptr_t)&s_bv[0][0];
    const uint32_t lds_ix = (uint32_t)(uintptr_t)&s_ix[0][0];
    const uint64_t bv_base = (uint64_t)(uintptr_t)bv;
    const uint64_t ix_base = (uint64_t)(uintptr_t)dst;

    float* forces = out + 1;
    float  esum   = 0.0f;

    int t = (int)blockIdx.x;
    const int stride = (int)gridDim.x;
    if (t < ntiles) {
        // ---- prefetch(tile, buf) as a lambda-free helper ----
        #define PREFETCH(TT, BB)                                                     \
            do {                                                                     \
                uint32_t _gb = (uint32_t)(TT) * (TILE * 12u);                        \
                uint32_t _gi = (uint32_t)(TT) * (TILE * 4u);                         \
                uint32_t _lb = lds_bv + (uint32_t)(BB) * (TILE * 12u);               \
                uint32_t _li = lds_ix + (uint32_t)(BB) * (TILE * 4u);                \
                _Pragma("unroll")                                                    \
                for (int _k = 0; _k < 3; ++_k) {                                     \
                    uint32_t _o = (threadIdx.x + (uint32_t)_k * TPB) * 16u;          \
                    async_b128(_lb + _o, _gb + _o, bv_base);                         \
                }                                                                    \
                async_b128(_li + threadIdx.x * 16u, _gi + threadIdx.x * 16u, ix_base);\
            } while (0)

        PREFETCH(t, 0);
        int buf = 0;
        while (t < ntiles) {
            int tn  = t + stride;
            int tpf = (tn < ntiles) ? tn : t;      // clamp: keeps async count uniform
            PREFETCH(tpf, buf ^ 1);
            wait_async_le();                        // current tile's 4 asyncs landed
            __syncthreads();

            #pragma unroll
            for (int k = 0; k < TILE / TPB; ++k) {
                int e = k * TPB + (int)threadIdx.x;
                float bx = s_bv[buf][e * 3 + 0];
                float by = s_bv[buf][e * 3 + 1];
                float bz = s_bv[buf][e * 3 + 2];
                int node = s_ix[buf][e];
                lj_edge(bx, by, bz, node, forces, esum);
            }
            __syncthreads();                        // all reads done before buf reuse
            t = tn;
            buf ^= 1;
        }
        wait_async_zero();
        #undef PREFETCH
    }
    energy_reduce_commit(esum, out);
}

// ---------------- tail edges (E % TILE), plain path --------------
__global__ void __launch_bounds__(TPB)
lj_tail(const float* __restrict__ bv, const int* __restrict__ dst,
        float* __restrict__ out, int start, int nedges) {
    int i = start + (int)(blockIdx.x * blockDim.x + threadIdx.x);
    float esum = 0.0f;
    if (i < nedges) {
        float bx = bv[3 * i + 0], by = bv[3 * i + 1], bz = bv[3 * i + 2];
        lj_edge(bx, by, bz, dst[i], out + 1, esum);
    }
    energy_reduce_commit(esum, out);
}

// ---------------- zero accumulators: out[0 .. 3N] --------------
__global__ void lj_zero(float* __restrict__ out, int n) {
    int i = (int)(blockIdx.x * blockDim.x + threadIdx.x);
    if (i < n) out[i] = 0.0f;
}

// ---------------- analytic_force = -2 * forces --------------
__global__ void lj_scale(float* __restrict__ out, int n3) {
    int i = (int)(blockIdx.x * blockDim.x + threadIdx.x);
    if (i < n3) out[1 + n3 + i] = -2.0f * out[1 + i];
}

extern "C" void kernel_launch(void* const* d_in, const int* in_sizes, int n_in,
                              void* d_out, int out_size, void* d_ws, size_t ws_size,
                              hipStream_t stream) {
    const float* bv  = (const float*)d_in[0];
    const int*   dst = (const int*)d_in[1];
    float* out = (float*)d_out;

    const int E  = in_sizes[0] / 3;        // 6,400,000
    const int N  = (out_size - 1) / 6;     // 100,000
    const int n3 = 3 * N;

    // 1) zero energy + force accumulators (d_out is poisoned by harness)
    {
        int n = n3 + 1;
        lj_zero<<<(n + TPB - 1) / TPB, TPB, 0, stream>>>(out, n);
    }

    // 2) main tiled pass with async global->LDS double buffering
    const int ntiles = E / TILE;
    if (ntiles > 0) {
        int grid = ntiles < 1024 ? ntiles : 1024;
        lj_main<<<grid, TPB, 0, stream>>>(bv, dst, out, ntiles);
    }

    // 3) tail edges (none for E = 6.4M, kept for generality)
    const int rem = E - ntiles * TILE;
    if (rem > 0) {
        lj_tail<<<(rem + TPB - 1) / TPB, TPB, 0, stream>>>(bv, dst, out, ntiles * TILE, E);
    }

    // 4) analytic_force = -2 * forces
    lj_scale<<<(n3 + TPB - 1) / TPB, TPB, 0, stream>>>(out, n3);
}